// GarchFilter_25262997635992
// MI455X (gfx1250) — compile-verified
//
#include <hip/hip_runtime.h>

// GARCH(1,1) particle filter scan for MI455X (gfx1250, wave32).
// One wave32 block per batch chain (grid = B = 128), adjacent particle pair
// (k = 2L, 2L+1) per lane (K = 64), sequential loop over T = 2048.
//
// Latency-oriented design (128 waves => the serial chain IS the runtime):
//  - 64-way sums via V_WMMA_F32_16X16X4_F32 (B = ones -> row sums), scalar
//    result extracted with two v_readlane (no LDS round trip).
//  - No logsumexp max shift: nv >= 1e-8 bounds loglik <= ~8.4, f32 exp safe;
//    underflow only zeroes negligible particles; mixture floor keeps CDF sane.
//  - CDF scan runs on UNNORMALIZED ex in parallel with the WMMA sum; the
//    normalization is absorbed into the search threshold (t' = u*S).
//  - Segmented 16-lane scan (4 ds_bpermute) + hoisted readlane for the half
//    total (pure cndmask-add, no exec branch).
//  - Resampling search: 7 readlane milestones (SGPR compares, SALU-float on
//    gfx1250) + two aligned ds_load_b128 -> one LDS round trip per draw.
//  - All divisions are raw v_rcp_f32.
// A separate kernel streams all of pf_out with aligned float4 NT stores; the
// scan kernel overwrites the h=4 slot of every record afterwards.

#define K_PART 64
#define H_DIM  5
#define TLEN   2048
#define BSZ    128

typedef __attribute__((ext_vector_type(2))) float v2f;
typedef __attribute__((ext_vector_type(4))) float f4;
typedef __attribute__((ext_vector_type(8))) float v8f;

__device__ __forceinline__ unsigned pcg(unsigned x) {
  x = x * 747796405u + 2891336453u;
  unsigned w = ((x >> ((x >> 28u) + 4u)) ^ x) * 277803737u;
  return (w >> 22u) ^ w;
}

__device__ __forceinline__ float u01(unsigned h) {
  return (float)(h >> 8) * (1.0f / 16777216.0f);
}

__device__ __forceinline__ float fastrcp(float x) {
  return __builtin_amdgcn_rcpf(x);
}

__device__ __forceinline__ float readlaneF(float x, int lane) {
  return __int_as_float(__builtin_amdgcn_readlane(__float_as_int(x), lane));
}

__device__ __forceinline__ float reduce64(float x0, float x1) {
#if __has_builtin(__builtin_amdgcn_wmma_f32_16x16x4_f32)
  v2f a;    a.x = x0;   a.y = x1;
  v2f ones; ones.x = 1.0f; ones.y = 1.0f;
  v8f c = {};
  c = __builtin_amdgcn_wmma_f32_16x16x4_f32(
      false, a, false, ones, (short)0, c, false, false);
  float s = ((c[0] + c[1]) + (c[2] + c[3])) + ((c[4] + c[5]) + (c[6] + c[7]));
  return readlaneF(s, 0) + readlaneF(s, 16);
#else
  float s = x0 + x1;
#pragma unroll
  for (int d = 16; d >= 1; d >>= 1) s += __shfl_xor(s, d, 32);
  return s;
#endif
}

__global__ void garch_fill_params(const float* __restrict__ gp,
                                  f4* __restrict__ pf4,
                                  long long nvec4) {
  __shared__ f4 s_pat[5];
  const float c0 = gp[0], c1 = gp[1], c2 = gp[2], c3 = gp[3];
  if (threadIdx.x < 5) {
    int m = threadIdx.x;
    f4 p;
#pragma unroll
    for (int j = 0; j < 4; ++j) {
      int e = m + j; e = (e >= 5) ? e - 5 : e;
      float v = (e == 0) ? c0 : (e == 1) ? c1 : (e == 2) ? c2 : (e == 3) ? c3 : 0.0f;
      p[j] = v;
    }
    s_pat[m] = p;
  }
  __syncthreads();

  long long f = (long long)blockIdx.x * blockDim.x + threadIdx.x;
  const long long stride = (long long)gridDim.x * blockDim.x;
  int m    = (int)((f * 4) % 5);
  const int step = (int)((stride * 4) % 5);
  for (; f < nvec4; f += stride) {
    f4 v = s_pat[m];
    __builtin_nontemporal_store(v, pf4 + f);
    m += step; m = (m >= 5) ? m - 5 : m;
  }
}

__global__ __launch_bounds__(32)
void garch_pf_scan(const float* __restrict__ obs,
                   const float* __restrict__ vol0,
                   const float* __restrict__ gp,
                   float* __restrict__ y_out,
                   float* __restrict__ pf_out) {
  const int b = blockIdx.x;
  const int L = threadIdx.x;
  const int k0 = 2 * L, k1 = 2 * L + 1;
  const float ef0 = (float)(k0 + 1);
  const float ef1 = (float)(k1 + 1);

  const float c0 = gp[0], g1 = gp[1], g2 = gp[2], g3 = gp[3];
  const float LOG2PI = 1.8378770664093454f;

  __shared__ __align__(16) float  s_cdf[K_PART];
  __shared__ float2               s_vr[K_PART];

  float v0 = vol0[(size_t)k0 * BSZ + b];
  float v1 = vol0[(size_t)k1 * BSZ + b];
  float w0 = 1.0f / 64.0f, w1 = 1.0f / 64.0f;

  const float* obs_b = obs + (size_t)b * TLEN * 2;
  float r_cur = obs_b[0];

  for (int t = 0; t < TLEN; ++t) {
    int tn = t + 1;   tn = (tn < TLEN) ? tn : TLEN - 1;
    int tp = t + 64;  tp = (tp < TLEN) ? tp : TLEN - 1;
    float r_next = obs_b[(size_t)tn * 2];
    __builtin_prefetch(obs_b + (size_t)tp * 2, 0, 1);

    const float r  = r_cur;
    const float r2 = r * r;
    const unsigned base = (unsigned)t * 2654435761u ^ (unsigned)b * 0x9E3779B9u;

    float ua0 = u01(pcg(base ^ (0x10000u + k0))) + 1e-12f;
    float ub0 = u01(pcg(base ^ (0x20000u + k0)));
    float ua1 = u01(pcg(base ^ (0x10000u + k1))) + 1e-12f;
    float ub1 = u01(pcg(base ^ (0x20000u + k1)));
    float e0 = __builtin_sqrtf(fmaxf(-2.0f * __logf(ua0), 0.0f)) * __cosf(6.28318530718f * ub0);
    float e1 = __builtin_sqrtf(fmaxf(-2.0f * __logf(ua1), 0.0f)) * __cosf(6.28318530718f * ub1);

    float nv0 = fabsf(c0 + g1 * v0 + g2 * r2 + 0.001f * g3 * e0) + 1e-8f;
    float nv1 = fabsf(c0 + g1 * v1 + g2 * r2 + 0.001f * g3 * e1) + 1e-8f;
    float ll0 = -0.5f * (r2 * fastrcp(nv0) + __logf(nv0) + LOG2PI);
    float ll1 = -0.5f * (r2 * fastrcp(nv1) + __logf(nv1) + LOG2PI);
    float ex0 = (w0 + 1e-12f) * __expf(ll0) + 1e-30f;
    float ex1 = (w1 + 1e-12f) * __expf(ll1) + 1e-30f;

    float S = reduce64(ex0, ex1);

    float sc = ex0 + ex1;
#pragma unroll
    for (int d = 1; d < 16; d <<= 1) {
      float y = __shfl_up(sc, d, 16);
      if ((L & 15) >= d) sc += y;
    }
    float lowTot = readlaneF(sc, 15);
    float scEx1 = sc + ((L >= 16) ? lowTot : 0.0f);
    float scEx0 = scEx1 - ex1;

    float cS   = (0.7f / 64.0f) * S;
    float cdf0 = 0.3f * scEx0 + cS * ef0;
    float cdf1 = 0.3f * scEx1 + cS * ef1;

    float rat0 = ex0 * fastrcp(0.3f * ex0 + cS);
    float rat1 = ex1 * fastrcp(0.3f * ex1 + cS);

    *(float2*)&s_cdf[k0] = make_float2(cdf0, cdf1);
    s_vr[k0] = make_float2(nv0, rat0);
    s_vr[k1] = make_float2(nv1, rat1);

    float m0 = readlaneF(cdf1, 3),  m1 = readlaneF(cdf1, 7);
    float m2 = readlaneF(cdf1, 11), m3 = readlaneF(cdf1, 15);
    float m4 = readlaneF(cdf1, 19), m5 = readlaneF(cdf1, 23);
    float m6 = readlaneF(cdf1, 27);
    __syncthreads();

    float t0 = u01(pcg(base ^ (0x30000u + k0))) * S;
    float t1 = u01(pcg(base ^ (0x30000u + k1))) * S;

    int ga = (m0 < t0) + (m1 < t0) + (m2 < t0) + (m3 < t0)
           + (m4 < t0) + (m5 < t0) + (m6 < t0);
    int gb = (m0 < t1) + (m1 < t1) + (m2 < t1) + (m3 < t1)
           + (m4 < t1) + (m5 < t1) + (m6 < t1);

    const f4* cdf4 = (const f4*)s_cdf;
    f4 caa = cdf4[2 * ga], cab = cdf4[2 * ga + 1];
    f4 cba = cdf4[2 * gb], cbb = cdf4[2 * gb + 1];
    int i0 = 8 * ga + (caa[0] < t0) + (caa[1] < t0) + (caa[2] < t0) + (caa[3] < t0)
                    + (cab[0] < t0) + (cab[1] < t0) + (cab[2] < t0);
    int i1 = 8 * gb + (cba[0] < t1) + (cba[1] < t1) + (cba[2] < t1) + (cba[3] < t1)
                    + (cbb[0] < t1) + (cbb[1] < t1) + (cbb[2] < t1);
    i0 = (i0 > 63) ? 63 : i0;
    i1 = (i1 > 63) ? 63 : i1;

    float2 gva = s_vr[i0];
    float2 gvb = s_vr[i1];
    float vr0 = gva.x, wr0 = gva.y;
    float vr1 = gvb.x, wr1 = gvb.y;

    float s1 = reduce64(wr0, wr1);
    float s2 = reduce64(vr0 * wr0, vr1 * wr1);
    float rs = fastrcp(s1);

    size_t rec0 = ((size_t)t * K_PART + k0) * BSZ + b;
    __builtin_nontemporal_store(vr0, pf_out + rec0 * H_DIM + 4);
    __builtin_nontemporal_store(vr1, pf_out + (rec0 + BSZ) * H_DIM + 4);
    if (L == 0) y_out[(size_t)t * BSZ + b] = s2 * rs;

    v0 = vr0; v1 = vr1; w0 = wr0 * rs; w1 = wr1 * rs;
    r_cur = r_next;
    __syncthreads();
  }
}

extern "C" void kernel_launch(void* const* d_in, const int* in_sizes, int n_in,
                              void* d_out, int out_size, void* d_ws, size_t ws_size,
                              hipStream_t stream) {
  const float* obs  = (const float*)d_in[0];
  const float* vol0 = (const float*)d_in[1];
  const float* gp   = (const float*)d_in[2];

  float* y_out  = (float*)d_out;
  float* pf_out = y_out + (size_t)TLEN * BSZ;

  const long long nvec4 = (long long)TLEN * K_PART * BSZ * H_DIM / 4;

  hipLaunchKernelGGL(garch_fill_params, dim3(2048), dim3(256), 0, stream,
                     gp, (f4*)pf_out, nvec4);
  hipLaunchKernelGGL(garch_pf_scan, dim3(BSZ), dim3(32), 0, stream,
                     obs, vol0, gp, y_out, pf_out);
}